// CRF_27943057227904
// MI455X (gfx1250) — compile-verified
//
#include <hip/hip_runtime.h>

typedef float v2f __attribute__((ext_vector_type(2)));
typedef float v8f __attribute__((ext_vector_type(8)));

#define KT        16
#define HDIM      1024
#define TLEN      1024
#define BATCH     64
#define START_TAG 14
#define STOP_TAG  15
#define NEGBIG    -3.0e38f

// ---------------------------------------------------------------------------
// Kernel 1: logits[r][n] = sum_h A[r][h]*W[h][n] + b[n], r in [0,65536)
// One wave per 16x16 output tile, K swept 4-at-a-time via V_WMMA_F32_16X16X4_F32.
// ---------------------------------------------------------------------------
__global__ __launch_bounds__(256)
void crf_gemm_wmma(const float* __restrict__ A,     // [B*T, H]
                   const float* __restrict__ W,     // [H, K]
                   const float* __restrict__ bias,  // [K]
                   float* __restrict__ out)         // [B*T, K]
{
    const int lane = threadIdx.x & 31;
    const int wave = threadIdx.x >> 5;
    const int tile = blockIdx.x * 8 + wave;        // 4096 tiles of 16 rows
    const int half = lane >> 4;                    // 0 or 1
    const int l    = lane & 15;

    // A: lane holds M = l, K-pair base = k0 + 2*half  (ISA 32-bit A 16x4 layout)
    const float* Arow = A + (size_t)(tile * 16 + l) * HDIM + 2 * half;
    // B: VGPR v holds row K = v + 2*half, col = l     (ISA 32-bit B 4x16 layout)
    const float* Wcol = W + (size_t)(2 * half) * KT + l;

    v8f acc0 = {0.f, 0.f, 0.f, 0.f, 0.f, 0.f, 0.f, 0.f};
    v8f acc1 = acc0;

#pragma unroll 4
    for (int k0 = 0; k0 < HDIM; k0 += 8) {
        v2f a0 = *(const v2f*)(Arow + k0);         // global_load_b64
        v2f a1 = *(const v2f*)(Arow + k0 + 4);
        v2f b0, b1;
        b0[0] = Wcol[(size_t)(k0    ) * KT];
        b0[1] = Wcol[(size_t)(k0 + 1) * KT];
        b1[0] = Wcol[(size_t)(k0 + 4) * KT];
        b1[1] = Wcol[(size_t)(k0 + 5) * KT];
        acc0 = __builtin_amdgcn_wmma_f32_16x16x4_f32(
            false, a0, false, b0, (short)0, acc0, false, false);
        acc1 = __builtin_amdgcn_wmma_f32_16x16x4_f32(
            false, a1, false, b1, (short)0, acc1, false, false);
    }

    v8f c = acc0 + acc1;
    const float bb = bias[l];
    // C/D layout: VGPR v -> row M = v + 8*half, col = l
    float* orow = out + (size_t)(tile * 16 + half * 8) * KT + l;
#pragma unroll
    for (int v = 0; v < 8; ++v)
        orow[(size_t)v * KT] = c[v] + bb;
}

// ---------------------------------------------------------------------------
// Kernel 2: per-sequence forward algorithm + gold path score.
// One wave (32 lanes) per batch element; lane j owns CRF state j (lanes 16-31
// mirror lanes 0-15). partial[b] = total_score(b) - real_path_score(b).
// ---------------------------------------------------------------------------
__global__ __launch_bounds__(32)
void crf_scan(const float* __restrict__ logits,  // [B, T, K]
              const float* __restrict__ trans,   // [K, K]
              const int*   __restrict__ tags,    // [B, T]
              float* __restrict__ partial)       // [B]
{
    const int b    = blockIdx.x;
    const int lane = threadIdx.x;
    const int j    = lane & 15;

    // column j of transitions: tcol[i] = trans[i][j]
    float tcol[16];
#pragma unroll
    for (int i = 0; i < 16; ++i) tcol[i] = trans[i * KT + j];

    const float* lg = logits + (size_t)b * TLEN * KT;

    float prev = 0.0f;   // reference scan init = zeros
    for (int t = 0; t < TLEN; ++t) {
        const float obs = lg[t * KT + j];
        float v[16];
        float m = NEGBIG;
#pragma unroll
        for (int i = 0; i < 16; ++i) {
            const float p = __shfl(prev, i, 32);
            v[i] = p + tcol[i];
            m = fmaxf(m, v[i]);
        }
        float s = 0.0f;
#pragma unroll
        for (int i = 0; i < 16; ++i) s += __expf(v[i] - m);
        // new[j] = obs_j + logsumexp_i(prev_i + trans[i][j])
        prev = obs + m + __logf(s);
    }

    // total = logsumexp_i(prev_i + trans[i][STOP])
    const float vstop = prev + trans[j * KT + STOP_TAG];
    float vi[16];
    float m = NEGBIG;
#pragma unroll
    for (int i = 0; i < 16; ++i) {
        vi[i] = __shfl(vstop, i, 32);
        m = fmaxf(m, vi[i]);
    }
    float s = 0.0f;
#pragma unroll
    for (int i = 0; i < 16; ++i) s += __expf(vi[i] - m);
    const float total = m + __logf(s);

    // gold path score: emissions + transitions along the tag sequence
    const int* tg = tags + (size_t)b * TLEN;
    float acc = 0.0f;
    for (int t = lane; t < TLEN; t += 32) {
        const int tag = tg[t];
        const int pt  = (t == 0) ? START_TAG : tg[t - 1];
        acc += lg[t * KT + tag] + trans[pt * KT + tag];
    }
#pragma unroll
    for (int off = 16; off > 0; off >>= 1)
        acc += __shfl_xor(acc, off, 32);

    if (lane == 0) {
        const float real = acc + trans[tg[TLEN - 1] * KT + STOP_TAG];
        partial[b] = total - real;
    }
}

// ---------------------------------------------------------------------------
// Kernel 3: deterministic reduction of 64 partials -> scalar loss.
// ---------------------------------------------------------------------------
__global__ __launch_bounds__(32)
void crf_reduce(const float* __restrict__ partial, float* __restrict__ out)
{
    const int lane = threadIdx.x;
    float s = partial[lane] + partial[lane + 32];
#pragma unroll
    for (int off = 16; off > 0; off >>= 1)
        s += __shfl_xor(s, off, 32);
    if (lane == 0) out[0] = s;
}

extern "C" void kernel_launch(void* const* d_in, const int* in_sizes, int n_in,
                              void* d_out, int out_size, void* d_ws, size_t ws_size,
                              hipStream_t stream)
{
    const float* lstm_out    = (const float*)d_in[0];  // [64,1024,1024]
    const float* W           = (const float*)d_in[1];  // [1024,16]
    const float* bias        = (const float*)d_in[2];  // [16]
    const float* transitions = (const float*)d_in[3];  // [16,16]
    const int*   tags        = (const int*)  d_in[4];  // [64,1024]

    float* logits  = (float*)d_ws;                        // 65536*16 floats = 4 MB
    float* partial = (float*)d_ws + (size_t)65536 * KT;   // 64 floats

    crf_gemm_wmma<<<512, 256, 0, stream>>>(lstm_out, W, bias, logits);
    crf_scan<<<BATCH, 32, 0, stream>>>(logits, transitions, tags, partial);
    crf_reduce<<<1, 32, 0, stream>>>(partial, (float*)d_out);
}